// MultiLevelSparseHashEncoding_18872086298817
// MI455X (gfx1250) — compile-verified
//
#include <hip/hip_runtime.h>
#include <stdint.h>

// ---------------- static config (mirrors the reference) ----------------
#define NLVL 16
#define NENC 16384            // power of two -> hash mod is a mask
#define TPB  256              // 8 wave32 waves per block
#define CHUNK_ELEMS (TPB * 3) // 768 floats of x per block (3 KB)

typedef unsigned int u32;
typedef unsigned int u32x4 __attribute__((ext_vector_type(4)));
typedef int          i32x4 __attribute__((ext_vector_type(4)));
typedef int          i32x8 __attribute__((ext_vector_type(8)));
typedef float        f32x2 __attribute__((ext_vector_type(2)));
typedef float        f32x4 __attribute__((ext_vector_type(4)));

// floor(16 * (512/16)^(l/15)) computed in float32 like the reference
__device__ constexpr int kRes[NLVL] = {
    16, 20, 25, 32, 40, 50, 64, 80, 101, 128, 161, 203, 256, 322, 406, 512};
// levels 0..2 have R^3 < 16384 -> direct ravel indexing; rest hashed.

__global__ __launch_bounds__(TPB) void
hashenc_kernel(const float* __restrict__ x, const float* __restrict__ emb,
               float* __restrict__ out, int npts)
{
    __shared__ float s_x[CHUNK_ELEMS];
    const int tid = threadIdx.x;
    const long long base_elem = (long long)blockIdx.x * CHUNK_ELEMS;

    // ---- stage this block's 256 points into LDS via the Tensor Data Mover ----
#if defined(__gfx1250__) && __has_builtin(__builtin_amdgcn_tensor_load_to_lds) && \
    __has_builtin(__builtin_amdgcn_s_wait_tensorcnt)
    if ((tid >> 5) == 0) {   // wave 0 only: TDM issues per-wave, EXEC ignored
        const uint64_t ga  = (uint64_t)(const void*)x + (uint64_t)base_elem * 4ull;
        const u32      lds = (u32)(size_t)(&s_x[0]);
        long long rem = (long long)npts * 3 - base_elem;
        u32 elems = (rem >= (long long)CHUNK_ELEMS) ? (u32)CHUNK_ELEMS
                    : (rem > 0 ? (u32)rem : 0u);

        // D# group 0: count=1 | lds_addr | global_addr[56:0] | type=2
        u32x4 g0;
        g0[0] = 1u;
        g0[1] = lds;
        g0[2] = (u32)ga;
        g0[3] = (u32)((ga >> 32) & 0x1FFFFFFu) | (2u << 30);

        // D# group 1: data_size=4B, 1-D tile: tile_dim0=768,
        // tensor_dim0=elems (OOB reads return zero for tail blocks),
        // tensor_dim1=1, tile_dim1=1, tile_dim2=0, dim0_stride=768.
        i32x8 g1;
        g1[0] = (int)(2u << 16);                        // data_size=4B
        g1[1] = (int)((elems & 0xFFFFu) << 16);         // tensor_dim0[15:0]
        g1[2] = (int)((elems >> 16) | (1u << 16));      // tensor_dim0 hi | tensor_dim1=1
        g1[3] = (int)((u32)CHUNK_ELEMS << 16);          // tile_dim0=768
        g1[4] = 1;                                      // tile_dim1=1, tile_dim2=0
        g1[5] = CHUNK_ELEMS;                            // tensor_dim0_stride lo
        g1[6] = 0;
        g1[7] = 0;

        i32x4 gz = {0, 0, 0, 0};
#if defined(__clang_major__) && (__clang_major__ >= 23)
        i32x8 gz8 = {0, 0, 0, 0, 0, 0, 0, 0};
        __builtin_amdgcn_tensor_load_to_lds(g0, g1, gz, gz, gz8, 0);
#else
        __builtin_amdgcn_tensor_load_to_lds(g0, g1, gz, gz, 0);
#endif
        __builtin_amdgcn_s_wait_tensorcnt(0);
    }
    __syncthreads();
#else
    for (int i = tid; i < CHUNK_ELEMS; i += TPB) {
        long long g = base_elem + i;
        s_x[i] = (g < (long long)npts * 3) ? x[g] : 0.0f;
    }
    __syncthreads();
#endif

    const int b = blockIdx.x * TPB + tid;
    if (b >= npts) return;

    // stride-3 LDS reads: gcd(3,64)=1 -> conflict-free across a wave32
    const float px0 = s_x[tid * 3 + 0];
    const float px1 = s_x[tid * 3 + 1];
    const float px2 = s_x[tid * 3 + 2];

    float o[2 * NLVL];

#pragma unroll
    for (int l = 0; l < NLVL; ++l) {
        const int  R      = kRes[l];
        const bool direct = (l < 3);
        // wave-uniform byte base of this level's table (saddr for gathers)
        const char* tblb  = (const char*)(emb + (size_t)l * (size_t)(NENC + 1) * 2u);
        const float hr    = 0.5f * (float)R;

        const float xs0 = (px0 + 1.0f) * hr - 0.5f;
        const float xs1 = (px1 + 1.0f) * hr - 0.5f;
        const float xs2 = (px2 + 1.0f) * hr - 0.5f;
        const float fl0 = floorf(xs0), fl1 = floorf(xs1), fl2 = floorf(xs2);
        const float fr0 = xs0 - fl0,   fr1 = xs1 - fl1,   fr2 = xs2 - fl2;
        const int   i0 = (int)fl0, i1 = (int)fl1, i2 = (int)fl2;

        // per-dim validity folded into the weights: any OOB dim variant zeroes
        // every corner product that uses it (== reference's zero sentinel row)
        const float w0[2] = {((u32)i0       < (u32)R) ? 1.0f - fr0 : 0.0f,
                             ((u32)(i0 + 1) < (u32)R) ? fr0        : 0.0f};
        const float w1[2] = {((u32)i1       < (u32)R) ? 1.0f - fr1 : 0.0f,
                             ((u32)(i1 + 1) < (u32)R) ? fr1        : 0.0f};
        const float w2[2] = {((u32)i2       < (u32)R) ? 1.0f - fr2 : 0.0f,
                             ((u32)(i2 + 1) < (u32)R) ? fr2        : 0.0f};
        const float w01[4] = {w0[0] * w1[0], w0[0] * w1[1],
                              w0[1] * w1[0], w0[1] * w1[1]};

        // per-dim index terms (2 variants each); hashed ids are in-bounds via
        // the mask, direct ids get per-dim clamps (value is dead when w==0)
        u32 t0[2], t1[2], t2[2];
        if (direct) {
            const int c0a = min(max(i0, 0), R - 1), c0b = min(max(i0 + 1, 0), R - 1);
            const int c1a = min(max(i1, 0), R - 1), c1b = min(max(i1 + 1, 0), R - 1);
            const int c2a = min(max(i2, 0), R - 1), c2b = min(max(i2 + 1, 0), R - 1);
            t0[0] = (u32)c0a * (u32)(R * R);  t0[1] = (u32)c0b * (u32)(R * R);
            t1[0] = (u32)c1a * (u32)R;        t1[1] = (u32)c1b * (u32)R;
            t2[0] = (u32)c2a;                 t2[1] = (u32)c2b;
        } else {
            t0[0] = (u32)i0;                  t0[1] = t0[0] + 1u;
            t1[0] = (u32)i1 * 2654435761u;    t1[1] = (u32)(i1 + 1) * 2654435761u;
            t2[0] = (u32)i2 * 805459861u;     t2[1] = (u32)(i2 + 1) * 805459861u;
        }

        // phase 1: 8 byte-offsets (32-bit) into the level table
        u32 off[8];
#pragma unroll
        for (int c = 0; c < 8; ++c) {
            const int b0 = (c >> 2) & 1, b1 = (c >> 1) & 1, b2 = c & 1;
            u32 id;
            if (direct) id = t0[b0] + t1[b1] + t2[b2];
            else        id = (t0[b0] ^ t1[b1] ^ t2[b2]) & (u32)(NENC - 1);
            off[c] = id << 3;                 // 8 bytes per row, fits in 32 bits
        }

        // phase 2: 8 gathers as uniform-base + 32-bit voffset (GVS saddr form)
        f32x2 ev[8];
#pragma unroll
        for (int c = 0; c < 8; ++c)
            ev[c] = *reinterpret_cast<const f32x2*>(tblb + (size_t)off[c]);

        // phase 3: reduce
        float acc0 = 0.0f, acc1 = 0.0f;
#pragma unroll
        for (int c = 0; c < 8; ++c) {
            const float w = w01[c >> 1] * w2[c & 1];
            acc0 = fmaf(w, ev[c][0], acc0);
            acc1 = fmaf(w, ev[c][1], acc1);
        }
        o[2 * l + 0] = acc0;
        o[2 * l + 1] = acc1;
    }

    // 128 B of contiguous output per point -> 8x b128 non-temporal stores
    f32x4* ov = reinterpret_cast<f32x4*>(out + (size_t)b * (2 * NLVL));
#pragma unroll
    for (int k = 0; k < 8; ++k) {
        f32x4 v = {o[4 * k + 0], o[4 * k + 1], o[4 * k + 2], o[4 * k + 3]};
        __builtin_nontemporal_store(v, ov + k);        // keep 128MB stream out of L2
    }
}

extern "C" void kernel_launch(void* const* d_in, const int* in_sizes, int n_in,
                              void* d_out, int out_size, void* d_ws, size_t ws_size,
                              hipStream_t stream) {
    (void)n_in; (void)out_size; (void)d_ws; (void)ws_size;
    const float* x   = (const float*)d_in[0];   // (B, 3) float32
    const float* emb = (const float*)d_in[1];   // (16, 16385, 2) float32
    float* out       = (float*)d_out;           // (B, 16, 2) float32
    const int npts   = in_sizes[0] / 3;
    const int blocks = (npts + TPB - 1) / TPB;
    hipLaunchKernelGGL(hashenc_kernel, dim3(blocks), dim3(TPB), 0, stream,
                       x, emb, out, npts);
}